// FNetBlock_73143293051491
// MI455X (gfx1250) — compile-verified
//
#include <hip/hip_runtime.h>
#include <hip/hip_bf16.h>

// ---------------------------------------------------------------------------
// FNet double-DCT2D == batched  out = A @ S @ B^T  over 1024 slices of 256x256
//   A = D(pi3) @ D(pi1),  B = D(pi2) @ D(pi0),  D(p)[k][n] = 2 cos(p k(2n+1)/512)
// Main kernel: fused per-slice GEMM using V_WMMA_F32_16X16X4_F32, slice staged
// in 320KB WGP LDS so x is read from HBM exactly once (~23us roofline).
// Round 4: pin the software pipeline with __builtin_amdgcn_sched_barrier so
// fragment ds_loads stay hoisted above the dependent WMMA (wait_dscnt > 0).
// ---------------------------------------------------------------------------

typedef __attribute__((ext_vector_type(2))) float v2f;
typedef __attribute__((ext_vector_type(8))) float v8f;

#define MAT_N    256
#define S_STRIDE 260   // padded row stride (floats): conflict-free + prefetch pad
#define B_STRIDE 260
#define T_STRIDE 17
#define T_ROWS   (MAT_N + 8)  // spare padded rows: depth-2 prefetch stays in-bounds
#define LDS_BYTES ((MAT_N * S_STRIDE + 16 * B_STRIDE + T_ROWS * T_STRIDE) * 4)

#if __has_builtin(__builtin_amdgcn_sched_barrier)
#define SCHED_FENCE() __builtin_amdgcn_sched_barrier(0)
#else
#define SCHED_FENCE()
#endif

// ----------------------------- threefry2x32-20 -----------------------------
__device__ __forceinline__ void threefry2x32(unsigned int k0, unsigned int k1,
                                             unsigned int c0, unsigned int c1,
                                             unsigned int& o0, unsigned int& o1) {
  const unsigned int ks2 = k0 ^ k1 ^ 0x1BD11BDAu;
  unsigned int ks[3] = {k0, k1, ks2};
  const unsigned int rotA[4] = {13u, 15u, 26u, 6u};
  const unsigned int rotB[4] = {17u, 29u, 16u, 24u};
  unsigned int x0 = c0 + k0;
  unsigned int x1 = c1 + k1;
#pragma unroll
  for (int blk = 0; blk < 5; ++blk) {
    const unsigned int* rot = (blk & 1) ? rotB : rotA;
#pragma unroll
    for (int r = 0; r < 4; ++r) {
      x0 += x1;
      x1 = (x1 << rot[r]) | (x1 >> (32u - rot[r]));
      x1 ^= x0;
    }
    const unsigned int s = (unsigned int)blk + 1u;
    x0 += ks[s % 3u];
    x1 += ks[(s + 1u) % 3u] + s;
  }
  o0 = x0;
  o1 = x1;
}

// --------------------------- tiny setup kernels ----------------------------
__global__ void zero_counts_kernel(unsigned int* counts) {
  if (threadIdx.x < 4) counts[threadIdx.x] = 0u;
}

__global__ void pi_mc_kernel(const int* __restrict__ pi_num,
                             unsigned int* __restrict__ counts) {
  const int est = blockIdx.y;                       // which of the 4 pi calls
  const unsigned int num = (unsigned int)pi_num[0];
  // fold_in(key(42), est)
  unsigned int k0, k1;
  threefry2x32(0u, 42u, 0u, (unsigned int)est, k0, k1);

  unsigned int local = 0u;
  const unsigned int stride = gridDim.x * blockDim.x;
  for (unsigned int j = blockIdx.x * blockDim.x + threadIdx.x; j < num; j += stride) {
    unsigned int r0, r1;
    threefry2x32(k0, k1, j, num + j, r0, r1);
    const float ux = __uint_as_float((r0 >> 9) | 0x3F800000u) - 1.0f;
    const float uy = __uint_as_float((r1 >> 9) | 0x3F800000u) - 1.0f;
    local += (ux * ux + uy * uy < 1.0f) ? 1u : 0u;
  }

  __shared__ unsigned int red[256];
  red[threadIdx.x] = local;
  __syncthreads();
  for (int s = 128; s > 0; s >>= 1) {
    if ((int)threadIdx.x < s) red[threadIdx.x] += red[threadIdx.x + s];
    __syncthreads();
  }
  if (threadIdx.x == 0) atomicAdd(&counts[est], red[0]);
}

__global__ void pi_finish_kernel(const int* __restrict__ pi_num,
                                 const unsigned int* __restrict__ counts,
                                 float* __restrict__ pis) {
  if (threadIdx.x < 4)
    pis[threadIdx.x] = 4.0f * (float)counts[threadIdx.x] / (float)pi_num[0];
}

// D_i[k][n] = 2 cos(pi_i * k * (2n+1) / 512)
__global__ void build_dct_kernel(const float* __restrict__ pis,
                                 float* __restrict__ D) {
  const int idx = blockIdx.x * blockDim.x + threadIdx.x;
  if (idx >= 4 * MAT_N * MAT_N) return;
  const int i = idx >> 16;
  const int k = (idx >> 8) & 255;
  const int n = idx & 255;
  const float p = pis[i];
  D[idx] = 2.0f * __cosf(p * (float)k * (float)(2 * n + 1) * (1.0f / 512.0f));
}

// A = D3 @ D1 ; B = D2 @ D0   (grid.y selects which)
__global__ void combine_kernel(const float* __restrict__ D,
                               float* __restrict__ A,
                               float* __restrict__ B) {
  const int i = blockIdx.x;
  const int j = threadIdx.x;
  const float* X;
  const float* Y;
  float* O;
  if (blockIdx.y == 0) { X = D + 3 * 65536; Y = D + 1 * 65536; O = A; }
  else                 { X = D + 2 * 65536; Y = D + 0 * 65536; O = B; }
  float s = 0.0f;
#pragma unroll 4
  for (int k = 0; k < MAT_N; ++k)
    s = fmaf(X[i * MAT_N + k], Y[k * MAT_N + j], s);
  O[i * MAT_N + j] = s;
}

// ------------------------ fused batched A @ S @ B^T ------------------------
// 1 workgroup = 1 slice.  512 threads = 16 waves; wave w owns output rows
// [16w, 16w+16).  Per 16-column block jb:
//   phase1: T = S @ B^T   (wmma, S/B from LDS, fenced 1-deep prefetch)
//   phase2: out = A @ T   (wmma, A register-resident, fenced 2-deep prefetch)
__global__ __launch_bounds__(512, 1)
void fused_dct2_gemm_kernel(const float* __restrict__ x,
                            const float* __restrict__ Amat,
                            const float* __restrict__ Bmat,
                            float* __restrict__ out) {
  extern __shared__ float lds[];
  float* Ssh = lds;                               // 256 x 260
  float* Bsh = lds + MAT_N * S_STRIDE;            // 16  x 260
  float* Tsh = Bsh + 16 * B_STRIDE;               // 264 x 17 (8 prefetch rows)

  const int tid  = threadIdx.x;
  const int wave = tid >> 5;
  const int lane = tid & 31;
  const int half = lane >> 4;   // 0/1: which k-pair / M-half per ISA layout
  const int lm   = lane & 15;   // fragment row (A) / col (B, D)

  const float* S = x + (size_t)blockIdx.x * (MAT_N * MAT_N);
  float*       O = out + (size_t)blockIdx.x * (MAT_N * MAT_N);

  // stage the whole slice into LDS (padded rows), float4 per transfer
  for (int idx = tid; idx < MAT_N * (MAT_N / 4); idx += 512) {
    const int row = idx >> 6;
    const int c4  = (idx & 63) << 2;
    const float4 v = *(const float4*)(S + row * MAT_N + c4);
    *(float4*)(Ssh + row * S_STRIDE + c4) = v;
  }

  // hoist this wave's A panel (16 rows x 256) into registers: it is reused
  // by all 16 jb blocks.  64 x v2f = 128 VGPRs per lane.
  const float* aRow = Amat + (wave * 16 + lm) * MAT_N + half * 2;
  v2f afrag[MAT_N / 4];
#pragma unroll
  for (int kk = 0; kk < MAT_N / 4; ++kk) {
    afrag[kk].x = aRow[4 * kk];
    afrag[kk].y = aRow[4 * kk + 1];
  }

  for (int jb = 0; jb < 16; ++jb) {
    __syncthreads();  // S staged / previous Tsh+Bsh consumers done
    // stage B rows [16jb, 16jb+16) (L2-resident, reused by all 1024 WGs)
    for (int idx = tid; idx < 16 * 64; idx += 512) {
      const int row = idx >> 6;
      const int c4  = (idx & 63) << 2;
      const float4 v = *(const float4*)(Bmat + (jb * 16 + row) * MAT_N + c4);
      *(float4*)(Bsh + row * B_STRIDE + c4) = v;
    }
    __syncthreads();

    // ---- phase 1: T[16w..16w+16, jb-block] = S @ B^T -----------------
    // fenced 1-deep prefetch: loads of step k+4 are pinned above the WMMA
    // consuming step k, so wait_dscnt only covers the previous iteration.
    // (final prefetch lands in row padding, value unused)
    v8f acc = {};
    const float* sRow = Ssh + (wave * 16 + lm) * S_STRIDE + half * 2;
    const float* bRow = Bsh + lm * B_STRIDE + half * 2;
    v2f a0, b0;
    a0.x = sRow[0]; a0.y = sRow[1];
    b0.x = bRow[0]; b0.y = bRow[1];
#pragma unroll 8
    for (int k = 0; k < MAT_N; k += 4) {
      v2f a1, b1;
      a1.x = sRow[k + 4]; a1.y = sRow[k + 5];
      b1.x = bRow[k + 4]; b1.y = bRow[k + 5];
      SCHED_FENCE();  // loads (k+4) may not sink below; wmma (k) may not hoist
      acc = __builtin_amdgcn_wmma_f32_16x16x4_f32(false, a0, false, b0,
                                                  (short)0, acc, false, false);
      a0 = a1; b0 = b1;
    }
#pragma unroll
    for (int v = 0; v < 8; ++v)
      Tsh[(wave * 16 + v + 8 * half) * T_STRIDE + lm] = acc[v];
    __syncthreads();

    // ---- phase 2: out[16w..16w+16, jb-block] = A @ T -----------------
    // fully unrolled so afrag[] stays register-resident; T fragments use a
    // fenced 2-deep rotation (prefetch lands in Tsh's spare padded rows)
    v8f acc2 = {};
    const float* tCol = Tsh + half * 2 * T_STRIDE + lm;
    v2f t0, t1;
    t0.x = tCol[0 * T_STRIDE];
    t0.y = tCol[1 * T_STRIDE];
    t1.x = tCol[4 * T_STRIDE];
    t1.y = tCol[5 * T_STRIDE];
#pragma unroll
    for (int kk = 0; kk < MAT_N / 4; ++kk) {
      v2f t2;
      t2.x = tCol[(4 * kk + 8) * T_STRIDE];
      t2.y = tCol[(4 * kk + 9) * T_STRIDE];
      SCHED_FENCE();  // load (kk+2) pinned above wmma (kk)
      acc2 = __builtin_amdgcn_wmma_f32_16x16x4_f32(false, afrag[kk], false, t0,
                                                   (short)0, acc2, false, false);
      t0 = t1; t1 = t2;
    }
#pragma unroll
    for (int v = 0; v < 8; ++v)
      O[(wave * 16 + v + 8 * half) * MAT_N + jb * 16 + lm] = acc2[v];
  }
}

// ------------------------------- launcher ----------------------------------
extern "C" void kernel_launch(void* const* d_in, const int* in_sizes, int n_in,
                              void* d_out, int out_size, void* d_ws, size_t ws_size,
                              hipStream_t stream) {
  (void)in_sizes; (void)n_in; (void)out_size; (void)ws_size;
  const float* x      = (const float*)d_in[0];
  const int*   pi_num = (const int*)d_in[1];
  float*       out    = (float*)d_out;

  // workspace layout
  unsigned int* counts = (unsigned int*)d_ws;                        // 16 B
  float* pis = (float*)((char*)d_ws + 16);                           // 16 B
  float* D   = (float*)((char*)d_ws + 64);                           // 1 MiB
  float* Ac  = D + 4 * MAT_N * MAT_N;                                // 256 KiB
  float* Bc  = Ac + MAT_N * MAT_N;                                   // 256 KiB

  // 1) four Monte-Carlo pi estimates (threefry2x32-20)
  zero_counts_kernel<<<1, 64, 0, stream>>>(counts);
  pi_mc_kernel<<<dim3(256, 4), 256, 0, stream>>>(pi_num, counts);
  pi_finish_kernel<<<1, 64, 0, stream>>>(pi_num, counts, pis);

  // 2) DCT matrices and their two compositions
  build_dct_kernel<<<(4 * MAT_N * MAT_N) / 256, 256, 0, stream>>>(pis, D);
  combine_kernel<<<dim3(MAT_N, 2), MAT_N, 0, stream>>>(D, Ac, Bc);

  // 3) fused batched  out = A @ S @ B^T  (1024 slices, ~301 KB LDS each)
  hipFuncSetAttribute((const void*)fused_dct2_gemm_kernel,
                      hipFuncAttributeMaxDynamicSharedMemorySize, LDS_BYTES);
  fused_dct2_gemm_kernel<<<1024, 512, LDS_BYTES, stream>>>(x, Ac, Bc, out);
}